// SkeletonConvolution_36318243455470
// MI455X (gfx1250) — compile-verified
//
#include <hip/hip_runtime.h>

// Problem constants (match reference setup_inputs)
#define NB   8
#define NC   64
#define NJ   22
#define NT   2048
#define NO   128
#define NK   7
#define KTOT 448            // NC*NK  : GEMM K dimension, kappa = (k<<6)|c
#define OT   16             // output-channel tile (WMMA M)
#define TT   64             // time tile per workgroup
#define BLK  128            // 4 waves of 32
#define SWS  452            // padded sW row stride (dwords) -> conflict-free b64 A loads
#define SXS  (TT + NK - 1)  // 70
#define SOS  65             // padded sO t-stride
#define SOI  (OT * SOS)     // 1040
#define WT_ELEMS ((size_t)NO * NC * NJ * NK)          // 1261568
#define WT_BYTES (WT_ELEMS * sizeof(float))           // ~5.05 MB

typedef __attribute__((ext_vector_type(2))) float v2f;
typedef __attribute__((ext_vector_type(8))) float v8f;

// ---- compile-time replication of find_neighbor(): column bitmasks of MASK ----
struct MaskBits { unsigned col[NJ]; };

constexpr MaskBits make_mask() {
  const int P[NJ]  = {0,0,1,2,3,0,5,6,7,0,9,10,11,10,13,14,15,10,17,18,19,12};
  const int EE[5]  = {4,8,16,20,21};
  int d[NJ][NJ] = {};
  for (int i = 0; i < NJ; i++)
    for (int j = 0; j < NJ; j++) d[i][j] = (i == j) ? 0 : 100000;
  for (int i = 0; i < NJ; i++) { int p = P[i]; if (d[i][p] > 1) { d[i][p] = 1; d[p][i] = 1; } }
  for (int k = 0; k < NJ; k++)
    for (int i = 0; i < NJ; i++)
      for (int j = 0; j < NJ; j++) {
        int v = d[i][k] + d[k][j];
        if (v < d[i][j]) d[i][j] = v;
      }
  bool nb[NJ][NJ] = {};
  for (int i = 0; i < NJ; i++)
    for (int j = 0; j < NJ; j++) nb[i][j] = (d[i][j] <= 2);
  bool gp[NJ] = {};
  for (int j = 0; j < NJ; j++) if (nb[0][j]) gp[j] = true;
  for (int e = 0; e < 5; e++) gp[EE[e]] = true;
  for (int j = 0; j < NJ; j++) if (gp[j]) nb[j][0] = true;
  for (int j = 0; j < NJ; j++) nb[0][j] = gp[j];
  MaskBits mb = {};
  for (int j = 0; j < NJ; j++) {
    unsigned m = 0;
    for (int i = 0; i < NJ; i++) if (nb[i][j]) m |= 1u << i;
    mb.col[j] = m;
  }
  return mb;
}

constexpr MaskBits MB = make_mask();

// One-time repack: Wt[j][o][(k<<6)|c] = W[o][c][j][k]  (exact LDS staging layout,
// so the hot loop streams 448 contiguous floats per row with full line utilization).
__global__ __launch_bounds__(256)
void repack_W(const float* __restrict__ W, float* __restrict__ Wt)
{
  size_t e = (size_t)blockIdx.x * 256 + threadIdx.x;
  if (e >= WT_ELEMS) return;
  int k = (int)(e % NK);
  int j = (int)((e / NK) % NJ);
  int c = (int)((e / ((size_t)NK * NJ)) % NC);
  int o = (int)(e / ((size_t)NK * NJ * NC));
  Wt[((size_t)j * NO + o) * KTOT + (k << 6) + c] = W[e];
}

// One workgroup: (b, o-tile of 16, t-tile of 64). Loops over all 22 joints,
// each wave computes a 16(o) x 16(t) WMMA tile of y_j over K=448 and
// scatter-adds it into the LDS out-accumulator rows i with MASK[i,j]==1.
__global__ __launch_bounds__(BLK)
void skel_conv_wmma(const float* __restrict__ x, const float* __restrict__ W,
                    const float* __restrict__ Wt, const int useWt,
                    const float* __restrict__ bias, float* __restrict__ out)
{
  __shared__ float sW[OT * SWS];     // W_j tile, kappa-major, padded rows
  __shared__ float sX[NC * SXS];     // x_j tile with halo, zero-padded at T edges
  __shared__ float sO[NJ * SOI];     // fused mask-mix accumulator

  const int tid = threadIdx.x;
  const int wv  = tid >> 5;          // wave id: t sub-tile
  const int ln  = tid & 31;
  const int n   = ln & 15;           // N column (t) / A row (o) within tile
  const int hi  = ln >> 4;           // half-wave select per ISA VGPR layouts
  const int t0  = blockIdx.x * TT;
  const int ot  = blockIdx.y * OT;
  const int b   = blockIdx.z;

  for (int e = tid; e < NJ * SOI; e += BLK) sO[e] = 0.0f;

  // A fragment base: A[m=n, K=2*hi + {0,1}] at kappa offset kb (immediate)
  const float* sArow = sW + n * SWS + 2 * hi;
  // B fragment base: B[K=kappa, N=n] = sX[c][wv*16 + n + k], c = (kb&63)+2*hi+q
  const float* sBrow = sX + 2 * hi * SXS + wv * 16 + n;

  for (int j = 0; j < NJ; ++j) {
    // ---- Stage W_j[16 x 448] -> sW[ol][(k<<6)|c] ----
    if (useWt) {
      // contiguous float4 stream from the repacked tensor
      const float4* src = (const float4*)(Wt + ((size_t)j * NO + ot) * KTOT);
      for (int e = tid; e < OT * (KTOT / 4); e += BLK) {
        int ol = e / (KTOT / 4);
        int q  = e - ol * (KTOT / 4);
        *(float4*)(sW + ol * SWS + (q << 2)) = src[e];
      }
    } else {
      // fallback: strided gather straight from W
      for (int e = tid; e < OT * KTOT; e += BLK) {
        int c  = e & 63;
        int kc = e >> 6;             // ol*7 + k
        int ol = kc / NK;
        int k  = kc - ol * NK;
        sW[ol * SWS + (k << 6) + c] =
            W[(((size_t)(ot + ol) * NC + c) * NJ + j) * NK + k];
      }
    }
    // ---- Stage x_j[64 x 70] with temporal halo of 3 each side ----
    for (int e = tid; e < NC * SXS; e += BLK) {
      int c  = e / SXS;
      int tt = e - c * SXS;
      int tg = t0 + tt - 3;
      float v = 0.0f;
      if ((unsigned)tg < (unsigned)NT)
        v = x[(((size_t)b * NC + c) * NJ + j) * NT + tg];
      sX[e] = v;
    }
    __syncthreads();

    // Warm L2 for the next joint while this one computes (global_prefetch_b8).
    if (j + 1 < NJ) {
      const int pc = tid & 63;
      const float* xn = x + (((size_t)b * NC + pc) * NJ + (j + 1)) * NT + t0;
      __builtin_prefetch(xn + ((tid >> 6) << 5), 0, 0);   // 2 lines per x row
      if (useWt) {
        const float* wn = Wt + ((size_t)(j + 1) * NO + ot) * KTOT;
        __builtin_prefetch(wn + tid * 32, 0, 0);          // 16 KB
        __builtin_prefetch(wn + 4096 + tid * 32, 0, 0);   // + 12 KB tail
      }
    }

    // 112 chained f32 WMMAs (2 independent accumulator chains for ILP).
    v8f acc0 = {};
    v8f acc1 = {};
#pragma unroll
    for (int kb = 0; kb < KTOT; kb += 4) {
      const int k  = kb >> 6;        // temporal tap (constant per step)
      const int cb = kb & 63;        // channel base (constant per step)
      v2f a = *(const v2f*)(sArow + kb);
      v2f bb;
      bb.x = sBrow[(cb + 0) * SXS + k];
      bb.y = sBrow[(cb + 1) * SXS + k];
      if ((kb >> 2) & 1)
        acc1 = __builtin_amdgcn_wmma_f32_16x16x4_f32(
            false, a, false, bb, (short)0, acc1, false, false);
      else
        acc0 = __builtin_amdgcn_wmma_f32_16x16x4_f32(
            false, a, false, bb, (short)0, acc0, false, false);
    }
    v8f acc = acc0 + acc1;

    // Fused skeleton-mask scatter: out_i += y_j for every i with MASK[i,j]=1.
    // C layout: VGPR r holds (o = r + 8*hi, t = wv*16 + n).
    {
      const unsigned colm = MB.col[j];
      float* p0 = sO + hi * 8 * SOS + wv * 16 + n;
#pragma unroll
      for (int i = 0; i < NJ; ++i) {
        if ((colm >> i) & 1u) {
          float* p = p0 + i * SOI;
#pragma unroll
          for (int r = 0; r < 8; ++r) p[r * SOS] += acc[r];
        }
      }
    }
    __syncthreads();   // protect sW/sX reuse next j; orders sO adds
  }

  // Write out[b, ot+o, i, t0+t] = sO + bias
  for (int e = tid; e < NJ * OT * TT; e += BLK) {
    int t = e & (TT - 1);
    int o = (e >> 6) & (OT - 1);
    int i = e >> 10;
    out[(((size_t)b * NO + ot + o) * NJ + i) * NT + t0 + t] =
        sO[i * SOI + o * SOS + t] + bias[ot + o];
  }
}

extern "C" void kernel_launch(void* const* d_in, const int* in_sizes, int n_in,
                              void* d_out, int out_size, void* d_ws, size_t ws_size,
                              hipStream_t stream) {
  (void)in_sizes; (void)n_in; (void)out_size;
  const float* x    = (const float*)d_in[0];
  const float* W    = (const float*)d_in[1];
  const float* bias = (const float*)d_in[2];
  float* out        = (float*)d_out;

  const int useWt = (ws_size >= WT_BYTES) ? 1 : 0;
  float* Wt = (float*)d_ws;
  if (useWt) {
    const int nblk = (int)((WT_ELEMS + 255) / 256);
    repack_W<<<nblk, 256, 0, stream>>>(W, Wt);
  }

  dim3 grid(NT / TT, NO / OT, NB);   // 32 x 8 x 8 = 2048 workgroups
  skel_conv_wmma<<<grid, BLK, 0, stream>>>(x, W, Wt, useWt, bias, out);
}